// DGCNN_seg_43602507989462
// MI455X (gfx1250) — compile-verified
//
#include <hip/hip_runtime.h>
#include <hip/hip_bf16.h>

typedef __attribute__((ext_vector_type(2))) float v2f;
typedef __attribute__((ext_vector_type(8))) float v8f;

#define NPTS 4096
#define NBATCH 4
#define KNBR 20
#define NEGS 0.2f
#define NCOLS (NPTS * KNBR) // 81920 columns per batch for edge convs

__device__ __forceinline__ v8f wmma4(v2f a, v2f b, v8f c) {
  // D = A(16x4,f32) x B(4x16,f32) + C(16x16,f32)
  return __builtin_amdgcn_wmma_f32_16x16x4_f32(false, a, false, b, (short)0, c,
                                               false, false);
}

__device__ __forceinline__ float lrelu(float v) { return v > 0.f ? v : NEGS * v; }

// ---------- small prep kernels ----------
__global__ void bnprep_kernel(const float* g, const float* bb, const float* m,
                              const float* v, float* S, float* T, int n) {
  int i = blockIdx.x * blockDim.x + threadIdx.x;
  if (i < n) {
    float s = g[i] / sqrtf(v[i] + 1e-5f);
    S[i] = s;
    T[i] = bb[i] - m[i] * s;
  }
}

// d[o][c] = w[o][C+c] - w[o][c], w is O x 2C row-major
__global__ void wdiff_kernel(const float* w, float* d, int O, int C) {
  int i = blockIdx.x * blockDim.x + threadIdx.x;
  if (i < O * C) {
    int o = i / C, c = i % C;
    d[i] = w[o * 2 * C + C + c] - w[o * 2 * C + c];
  }
}

// U1[b][o][k] = sum_{c<5} w1[o][c] * x[b][c][k], k<20
__global__ void u1_kernel(const float* x, const float* w1, float* U1) {
  int i = blockIdx.x * blockDim.x + threadIdx.x;
  if (i < NBATCH * 64 * KNBR) {
    int k = i % KNBR, o = (i / KNBR) % 64, b = i / (KNBR * 64);
    const float* xb = x + (long)b * 5 * NPTS;
    float s = 0.f;
    for (int c = 0; c < 5; c++) s += w1[o * 10 + c] * xb[c * NPTS + k];
    U1[i] = s;
  }
}

// T1[b][o][n] = sum_{c<5} w1d[o][c] * x[b][c][n]
__global__ void t1_kernel(const float* x, const float* w1d, float* T1) {
  int i = blockIdx.x * blockDim.x + threadIdx.x;
  if (i < NBATCH * 64 * NPTS) {
    int n = i % NPTS, o = (i / NPTS) % 64, b = i / (64 * NPTS);
    const float* xb = x + (long)b * 5 * NPTS;
    float s = 0.f;
    for (int c = 0; c < 5; c++) s += w1d[o * 5 + c] * xb[c * NPTS + n];
    T1[i] = s;
  }
}

// ---------- edge-feature materialization ----------
// stage1: neighbor j == k (top_k over all-zero distances -> [0..19])
__global__ void makey1_kernel(const float* U1b, const float* T1b,
                              const float* S, const float* T, float* Y) {
  int i = blockIdx.x * blockDim.x + threadIdx.x;
  if (i < 64 * NCOLS) {
    int k = i % KNBR, n = (i / KNBR) % NPTS, c = i / (KNBR * NPTS);
    float v = U1b[c * KNBR + k] + T1b[c * NPTS + n];
    Y[i] = lrelu(v * S[c] + T[c]);
  }
}

__global__ void makey3_kernel(const float* Ub, const float* Tb, const int* idxb,
                              const float* S, const float* T, float* Y) {
  int i = blockIdx.x * blockDim.x + threadIdx.x;
  if (i < 64 * NCOLS) {
    int k = i % KNBR, n = (i / KNBR) % NPTS, c = i / (KNBR * NPTS);
    int j = idxb[n * KNBR + k];
    float v = Ub[c * NPTS + j] + Tb[c * NPTS + n];
    Y[i] = lrelu(v * S[c] + T[c]);
  }
}

// ---------- generic WMMA fp32 GEMM, Out = act(bn(W x In)) ----------
// W: M x C row-major with leading dim ldW; In: C x L; Out: M x L.
// One wave (32 threads) computes a 16(M) x 64(N) tile: 4 accumulators,
// each A (weight) fragment feeds 4 v_wmma -> 4x weight-operand reuse.
// L must be a multiple of 64, C a multiple of 4.
__global__ void gemm_wmma_kernel(const float* __restrict__ W, int ldW,
                                 const float* __restrict__ In, long inStride,
                                 float* __restrict__ Out, long outStride,
                                 const float* __restrict__ S,
                                 const float* __restrict__ T, int C, int L,
                                 int act) {
  const float* inb = In + (long)blockIdx.z * inStride;
  float* outb = Out + (long)blockIdx.z * outStride;
  int o0 = blockIdx.y * 16, col0 = blockIdx.x * 64;
  int tid = threadIdx.x, lo = tid & 15, hi = tid >> 4;
  v8f acc0 = (v8f)(0.0f);
  v8f acc1 = acc0, acc2 = acc0, acc3 = acc0;
  const float* wr = W + (long)(o0 + lo) * ldW + hi * 2;
  const float* ir = inb + (long)(hi * 2) * L + col0 + lo;
  for (int kk = 0; kk < C; kk += 4) {
    v2f a;
    a.x = wr[0];
    a.y = wr[1];
    v2f b0, b1, b2, b3;
    b0.x = ir[0];
    b0.y = ir[L];
    b1.x = ir[16];
    b1.y = ir[L + 16];
    b2.x = ir[32];
    b2.y = ir[L + 32];
    b3.x = ir[48];
    b3.y = ir[L + 48];
    acc0 = wmma4(a, b0, acc0);
    acc1 = wmma4(a, b1, acc1);
    acc2 = wmma4(a, b2, acc2);
    acc3 = wmma4(a, b3, acc3);
    wr += 4;
    ir += 4L * L;
  }
  v8f accs[4] = {acc0, acc1, acc2, acc3};
#pragma unroll
  for (int r = 0; r < 8; r++) {
    int m = o0 + r + 8 * hi;
    float sm = S ? S[m] : 1.f;
    float tm = S ? T[m] : 0.f;
    float* op = outb + (long)m * L + col0 + lo;
#pragma unroll
    for (int t = 0; t < 4; t++) {
      float v = accs[t][r];
      v = v * sm + tm;
      if (act) v = lrelu(v);
      op[t * 16] = v;
    }
  }
}

// ---------- reductions / gathers ----------
__global__ void kmax_kernel(const float* Y, float* Xo) {
  int i = blockIdx.x * blockDim.x + threadIdx.x;
  if (i < 64 * NPTS) {
    int n = i % NPTS, c = i / NPTS;
    const float* p = Y + (long)c * NCOLS + n * KNBR;
    float mx = p[0];
    for (int k = 1; k < KNBR; k++) mx = fmaxf(mx, p[k]);
    Xo[i] = mx;
  }
}

__global__ void sq_kernel(const float* X, float* sq) {
  int i = blockIdx.x * blockDim.x + threadIdx.x;
  if (i < NBATCH * NPTS) {
    int n = i % NPTS, b = i / NPTS;
    const float* xb = X + (long)b * 64 * NPTS;
    float s = 0.f;
    for (int c = 0; c < 64; c++) {
      float t = xb[c * NPTS + n];
      s += t * t;
    }
    sq[i] = s;
  }
}

// Fused Gram (V_WMMA_F32_16X16X4_F32) + top-20 per row.
// pd[m][n] = 2*x_m.x_n - |x_m|^2 - |x_n|^2 ; keep 20 largest, stable ties.
// One wave per 16 rows; 64 columns per sweep (4 accumulators, A reused 4x).
__global__ void knn_kernel(const float* X, const float* sq, int* idx) {
  __shared__ float s_tile[16 * 64];
  __shared__ float s_topv[16 * KNBR];
  __shared__ int s_topi[16 * KNBR];
  int b = blockIdx.y;
  int m0 = blockIdx.x * 16;
  int tid = threadIdx.x, lo = tid & 15, hi = tid >> 4;
  const float* Xb = X + (long)b * 64 * NPTS;
  const float* sqb = sq + b * NPTS;
  v2f areg[16];
#pragma unroll
  for (int kk = 0; kk < 16; kk++) {
    areg[kk].x = Xb[(kk * 4 + hi * 2) * NPTS + m0 + lo];
    areg[kk].y = Xb[(kk * 4 + hi * 2 + 1) * NPTS + m0 + lo];
  }
  float sqm[8];
#pragma unroll
  for (int r = 0; r < 8; r++) sqm[r] = sqb[m0 + r + 8 * hi];
  for (int i = tid; i < 16 * KNBR; i += 32) {
    s_topv[i] = -3.0e38f;
    s_topi[i] = 0;
  }
  __syncthreads();
  for (int nt = 0; nt < NPTS / 64; nt++) {
    int n0 = nt * 64;
    v8f acc0 = (v8f)(0.0f);
    v8f acc1 = acc0, acc2 = acc0, acc3 = acc0;
    const float* ir = Xb + (long)(hi * 2) * NPTS + n0 + lo;
#pragma unroll
    for (int kk = 0; kk < 16; kk++) {
      v2f b0, b1, b2, b3;
      b0.x = ir[0];
      b0.y = ir[NPTS];
      b1.x = ir[16];
      b1.y = ir[NPTS + 16];
      b2.x = ir[32];
      b2.y = ir[NPTS + 32];
      b3.x = ir[48];
      b3.y = ir[NPTS + 48];
      acc0 = wmma4(areg[kk], b0, acc0);
      acc1 = wmma4(areg[kk], b1, acc1);
      acc2 = wmma4(areg[kk], b2, acc2);
      acc3 = wmma4(areg[kk], b3, acc3);
      ir += 4L * NPTS;
    }
    v8f accs[4] = {acc0, acc1, acc2, acc3};
#pragma unroll
    for (int t = 0; t < 4; t++) {
      float sqn = sqb[n0 + t * 16 + lo];
#pragma unroll
      for (int r = 0; r < 8; r++)
        s_tile[(r + 8 * hi) * 64 + t * 16 + lo] =
            2.f * accs[t][r] - sqm[r] - sqn;
    }
    __syncthreads();
    if (tid < 16) {
      float* tv = s_topv + tid * KNBR;
      int* ti = s_topi + tid * KNBR;
      for (int j = 0; j < 64; j++) { // ascending index => stable ties
        float val = s_tile[tid * 64 + j];
        if (val > tv[KNBR - 1]) {
          int p = KNBR - 1;
          while (p > 0 && tv[p - 1] < val) {
            tv[p] = tv[p - 1];
            ti[p] = ti[p - 1];
            p--;
          }
          tv[p] = val;
          ti[p] = n0 + j;
        }
      }
    }
    __syncthreads();
  }
  if (tid < 16) {
    int n = m0 + tid;
    for (int k = 0; k < KNBR; k++)
      idx[((long)b * NPTS + n) * KNBR + k] = s_topi[tid * KNBR + k];
  }
}

// stage D: x3[b][c][n] = max_k lrelu(bn5(U[c][j] + T[c][n]))
__global__ void edgemax_kernel(const float* U, const float* T, const int* idx,
                               const float* S, const float* Tb, float* Xo) {
  int i = blockIdx.x * blockDim.x + threadIdx.x;
  if (i < NBATCH * 64 * NPTS) {
    int n = i % NPTS, c = (i / NPTS) % 64, b = i / (64 * NPTS);
    const float* Ub = U + (long)b * 64 * NPTS;
    const int* ib = idx + ((long)b * NPTS + n) * KNBR;
    float t = T[(long)b * 64 * NPTS + c * NPTS + n];
    float s = S[c], bb = Tb[c];
    float mx = -3.0e38f;
    for (int k = 0; k < KNBR; k++) {
      int j = ib[k];
      float v = (Ub[c * NPTS + j] + t) * s + bb;
      mx = fmaxf(mx, lrelu(v));
    }
    Xo[i] = mx;
  }
}

__global__ void concat192_kernel(const float* x1, const float* x2,
                                 const float* x3, float* O) {
  int i = blockIdx.x * blockDim.x + threadIdx.x;
  if (i < NBATCH * 192 * NPTS) {
    int n = i % NPTS, c = (i / NPTS) % 192, b = i / (192 * NPTS);
    const float* src = c < 64 ? x1 : (c < 128 ? x2 : x3);
    O[i] = src[((long)b * 64 + (c & 63)) * NPTS + n];
  }
}

__global__ void concat256_kernel(const float* gm, const float* x1,
                                 const float* x2, const float* x3, float* O) {
  int i = blockIdx.x * blockDim.x + threadIdx.x;
  if (i < NBATCH * 256 * NPTS) {
    int n = i % NPTS, c = (i / NPTS) % 256, b = i / (256 * NPTS);
    float v;
    if (c < 64)
      v = gm[b * 64 + c];
    else {
      const float* src = c < 128 ? x1 : (c < 192 ? x2 : x3);
      v = src[((long)b * 64 + (c & 63)) * NPTS + n];
    }
    O[i] = v;
  }
}

__global__ void gmax_kernel(const float* h6, float* gm) {
  __shared__ float red[256];
  int bc = blockIdx.x;
  const float* p = h6 + (long)bc * NPTS;
  float mx = -3.0e38f;
  for (int n = threadIdx.x; n < NPTS; n += 256) mx = fmaxf(mx, p[n]);
  red[threadIdx.x] = mx;
  __syncthreads();
  for (int s = 128; s > 0; s >>= 1) {
    if (threadIdx.x < s)
      red[threadIdx.x] = fmaxf(red[threadIdx.x], red[threadIdx.x + s]);
    __syncthreads();
  }
  if (threadIdx.x == 0) gm[bc] = red[0];
}

__global__ void out9_kernel(const float* w9, const float* h8, float* out) {
  int i = blockIdx.x * blockDim.x + threadIdx.x;
  if (i < NBATCH * NPTS) {
    int n = i % NPTS, b = i / NPTS;
    const float* hb = h8 + (long)b * 256 * NPTS;
    float s = 0.f;
    for (int c = 0; c < 256; c++) s += w9[c] * hb[c * NPTS + n];
    out[i] = s;
  }
}

static inline int cdiv(long a, int b) { return (int)((a + b - 1) / b); }

extern "C" void kernel_launch(void* const* d_in, const int* in_sizes, int n_in,
                              void* d_out, int out_size, void* d_ws,
                              size_t ws_size, hipStream_t stream) {
  const float* x = (const float*)d_in[0];
  const float* w[9];
  const float *bg[8], *bb[8], *bm[8], *bv[8];
  for (int i = 0; i < 8; i++) {
    w[i] = (const float*)d_in[1 + 5 * i];
    bg[i] = (const float*)d_in[2 + 5 * i];
    bb[i] = (const float*)d_in[3 + 5 * i];
    bm[i] = (const float*)d_in[4 + 5 * i];
    bv[i] = (const float*)d_in[5 + 5 * i];
  }
  const float* w9p = (const float*)d_in[41];
  float* out = (float*)d_out;

  float* P = (float*)d_ws;
  const long MB = 262144; // floats per MiB
  // params sub-offsets (floats)
  float* w1d = P + 0;    // 320
  float* w3d = P + 512;  // 4096
  float* w5d = P + 4608; // 4096
  float *S[8], *T[8];
  for (int i = 0; i < 6; i++) {
    S[i] = P + 8704 + i * 128;
    T[i] = S[i] + 64;
  }
  S[6] = P + 9472;
  T[6] = P + 9984; // 512 each
  S[7] = P + 10496;
  T[7] = P + 10752;      // 256 each
  float* U1 = P + 11008; // B*64*20
  // big buffers (MiB offsets, with aliasing)
  float* sqb = P + 1 * MB;
  int* idxb = (int*)(P + 2 * MB); // B*N*K ints
  float* x1 = P + 4 * MB;
  float* x2 = P + 8 * MB;
  float* x3 = P + 12 * MB;
  float* Ub = P + 16 * MB;
  float* Tb = P + 20 * MB;
  float* h6 = P + 24 * MB;
  float* Y1 = P + 28 * MB; // 20 MiB; aliased later by In192
  float* Y2 = P + 48 * MB; // 20 MiB; aliased later by In256, then h8
  float* In192 = P + 28 * MB;
  float* In256 = P + 48 * MB;
  float* h7 = P + 68 * MB;     // 32 MiB
  float* h8 = P + 48 * MB;     // reuse after In256 consumed
  float* gmaxb = P + 100 * MB; // 256 floats

  const long cN = 64L * NPTS;
  dim3 blk(256);

  // ---- derived weights + BN affine folding ----
  wdiff_kernel<<<cdiv(320, 256), blk, 0, stream>>>(w[0], w1d, 64, 5);
  wdiff_kernel<<<cdiv(4096, 256), blk, 0, stream>>>(w[2], w3d, 64, 64);
  wdiff_kernel<<<cdiv(4096, 256), blk, 0, stream>>>(w[4], w5d, 64, 64);
  const int bncnt[8] = {64, 64, 64, 64, 64, 64, 512, 256};
  for (int i = 0; i < 8; i++)
    bnprep_kernel<<<cdiv(bncnt[i], 256), blk, 0, stream>>>(
        bg[i], bb[i], bm[i], bv[i], S[i], T[i], bncnt[i]);
  // ---- stage A: graph feature on empty slice => neighbors = [0..19] ----
  u1_kernel<<<cdiv(NBATCH * 64 * KNBR, 256), blk, 0, stream>>>(x, w[0], U1);
  t1_kernel<<<cdiv(NBATCH * cN, 256), blk, 0, stream>>>(x, w1d, Tb);
  for (int b = 0; b < NBATCH; b++) {
    makey1_kernel<<<cdiv(64L * NCOLS, 256), blk, 0, stream>>>(
        U1 + b * 64 * KNBR, Tb + b * cN, S[0], T[0], Y1);
    gemm_wmma_kernel<<<dim3(NCOLS / 64, 4, 1), 32, 0, stream>>>(
        w[1], 64, Y1, 0, Y2, 0, S[1], T[1], 64, NCOLS, 1);
    kmax_kernel<<<cdiv(cN, 256), blk, 0, stream>>>(Y2, x1 + b * cN);
  }
  // ---- stage B/C: knn(x1), edge conv W3 (decomposed) + W4 ----
  sq_kernel<<<cdiv(NBATCH * NPTS, 256), blk, 0, stream>>>(x1, sqb);
  knn_kernel<<<dim3(NPTS / 16, NBATCH), 32, 0, stream>>>(x1, sqb, idxb);
  gemm_wmma_kernel<<<dim3(NPTS / 64, 4, NBATCH), 32, 0, stream>>>(
      w[2], 128, x1, cN, Ub, cN, nullptr, nullptr, 64, NPTS, 0);
  gemm_wmma_kernel<<<dim3(NPTS / 64, 4, NBATCH), 32, 0, stream>>>(
      w3d, 64, x1, cN, Tb, cN, nullptr, nullptr, 64, NPTS, 0);
  for (int b = 0; b < NBATCH; b++) {
    makey3_kernel<<<cdiv(64L * NCOLS, 256), blk, 0, stream>>>(
        Ub + b * cN, Tb + b * cN, idxb + (long)b * NPTS * KNBR, S[2], T[2], Y1);
    gemm_wmma_kernel<<<dim3(NCOLS / 64, 4, 1), 32, 0, stream>>>(
        w[3], 64, Y1, 0, Y2, 0, S[3], T[3], 64, NCOLS, 1);
    kmax_kernel<<<cdiv(cN, 256), blk, 0, stream>>>(Y2, x2 + b * cN);
  }
  // ---- stage D: knn(x2), edge conv W5 (decomposed), fused max ----
  sq_kernel<<<cdiv(NBATCH * NPTS, 256), blk, 0, stream>>>(x2, sqb);
  knn_kernel<<<dim3(NPTS / 16, NBATCH), 32, 0, stream>>>(x2, sqb, idxb);
  gemm_wmma_kernel<<<dim3(NPTS / 64, 4, NBATCH), 32, 0, stream>>>(
      w[4], 128, x2, cN, Ub, cN, nullptr, nullptr, 64, NPTS, 0);
  gemm_wmma_kernel<<<dim3(NPTS / 64, 4, NBATCH), 32, 0, stream>>>(
      w5d, 64, x2, cN, Tb, cN, nullptr, nullptr, 64, NPTS, 0);
  edgemax_kernel<<<cdiv(NBATCH * cN, 256), blk, 0, stream>>>(Ub, Tb, idxb,
                                                             S[4], T[4], x3);
  // ---- head: W6, global max, W7, W8, W9 ----
  concat192_kernel<<<cdiv(NBATCH * 192L * NPTS, 256), blk, 0, stream>>>(
      x1, x2, x3, In192);
  gemm_wmma_kernel<<<dim3(NPTS / 64, 4, NBATCH), 32, 0, stream>>>(
      w[5], 192, In192, 192L * NPTS, h6, cN, S[5], T[5], 192, NPTS, 1);
  gmax_kernel<<<NBATCH * 64, 256, 0, stream>>>(h6, gmaxb);
  concat256_kernel<<<cdiv(NBATCH * 256L * NPTS, 256), blk, 0, stream>>>(
      gmaxb, x1, x2, x3, In256);
  gemm_wmma_kernel<<<dim3(NPTS / 64, 32, NBATCH), 32, 0, stream>>>(
      w[6], 256, In256, 256L * NPTS, h7, 512L * NPTS, S[6], T[6], 256, NPTS, 1);
  gemm_wmma_kernel<<<dim3(NPTS / 64, 16, NBATCH), 32, 0, stream>>>(
      w[7], 512, h7, 512L * NPTS, h8, 256L * NPTS, S[7], T[7], 512, NPTS, 1);
  out9_kernel<<<cdiv(NBATCH * NPTS, 256), blk, 0, stream>>>(w9p, h8, out);
}